// SimpleJastrow_90795608637680
// MI455X (gfx1250) — compile-verified
//
#include <hip/hip_runtime.h>

typedef float v2f __attribute__((ext_vector_type(2)));
typedef float v8f __attribute__((ext_vector_type(8)));

// One workgroup per batch (B=4096 blocks). 128 threads = 4 wave32 waves.
// Wave w owns row-block w (16 rows) of the 64x64 Gram matrix G = X·Xᵀ
// (X = [pos.xyz, spin], so G = pos·pos + s_i·s_j) and computes it with
// V_WMMA_F32_16X16X4_F32 against all 4 column blocks (fully unrolled,
// uniform control flow). The strict lower-triangle mask i>j is applied
// branchlessly per element; it also zeroes whole above-diagonal tiles.
__global__ __launch_bounds__(128) void simple_jastrow_wmma(
    const float* __restrict__ x,        // [B, 64, 4]
    const float* __restrict__ alpha_parallel,
    const float* __restrict__ alpha_anti,
    float* __restrict__ out)            // [B]
{
    __shared__ __align__(16) float4 xs[64];   // staged batch tile (1 KiB)
    __shared__ __align__(16) float2 ns[64];   // {|pos_i|^2, spin_i}
    __shared__ float partial[4];

    const int b    = blockIdx.x;
    const int tid  = threadIdx.x;
    const int wave = tid >> 5;     // 0..3
    const int lane = tid & 31;

    // ---- stage batch into LDS, precompute norms/spins (read input once) ----
    const float4* xb = reinterpret_cast<const float4*>(x) + (size_t)b * 64;
    if (tid < 64) {
        float4 v = xb[tid];
        xs[tid] = v;
        ns[tid] = make_float2(v.x * v.x + v.y * v.y + v.z * v.z, v.w);
    }
    __syncthreads();

    const float ap    = alpha_parallel[0];
    const float aa    = alpha_anti[0];
    const float cpar  = 0.25f * ap * ap;   // weight folded into numerator
    const float canti = 0.50f * aa * aa;

    const int i0    = wave * 16;
    const int lrow  = lane & 15;
    const int hi    = lane >> 4;           // A: K={0,1} vs {2,3};  C/D: M bias +8
    const int ibase = i0 + 8 * hi;         // this lane's 8 row indices: ibase..ibase+7

    const float* xsf = reinterpret_cast<const float*>(xs);

    // A fragment (16x4 f32): lane<16 -> x[i0+lane].xy ; lane>=16 -> x[i0+lane-16].zw
    const v2f afrag = *reinterpret_cast<const v2f*>(&xsf[(i0 + lrow) * 4 + 2 * hi]);

    // Hoist the 8 row-side {norm, spin} pairs (contiguous, aligned LDS reads)
    float ni[8], si[8];
#pragma unroll
    for (int r = 0; r < 8; ++r) {
        float2 t = ns[ibase + r];
        ni[r] = t.x;
        si[r] = t.y;
    }

    float acc = 0.0f;

#pragma unroll
    for (int jblk = 0; jblk < 4; ++jblk) {
        const int j0 = jblk * 16;
        const int j  = j0 + lrow;

        // B fragment (4x16 f32): mirrored layout, N in lanes
        const v2f bfrag = *reinterpret_cast<const v2f*>(&xsf[j * 4 + 2 * hi]);

        v8f c = {};
        c = __builtin_amdgcn_wmma_f32_16x16x4_f32(
                /*neg_a=*/false, afrag,
                /*neg_b=*/false, bfrag,
                /*c_mod=*/(short)0, c,
                /*reuse_a=*/false, /*reuse_b=*/false);

        const float2 nsj = ns[j];
        const float  nj  = nsj.x;
        const float  sj  = nsj.y;

#pragma unroll
        for (int r = 0; r < 8; ++r) {
            const int   i   = ibase + r;        // C/D layout: VGPR r -> M = r + 8*hi
            const float ss  = si[r] * sj;       // +1 parallel, -1 anti
            const float dot = c[r] - ss;        // position-only dot product
            float sq = fmaf(-2.0f, dot, ni[r] + nj);
            sq = fmaxf(sq, 0.0f);
            const float d    = __builtin_amdgcn_sqrtf(sq);
            const bool  same = ss > 0.0f;
            const float num  = same ? cpar : canti;
            const float den  = (same ? ap : aa) + d;
            const float contrib = num * __builtin_amdgcn_rcpf(den);
            acc += (i > j) ? contrib : 0.0f;    // branchless strict lower triangle
        }
    }

    // ---- deterministic reduction: wave32 shuffle, then fixed-order combine ----
#pragma unroll
    for (int off = 16; off > 0; off >>= 1)
        acc += __shfl_down(acc, off, 32);

    if (lane == 0) partial[wave] = acc;
    __syncthreads();

    if (tid == 0) {
        const float s = ((partial[0] + partial[1]) + partial[2]) + partial[3];
        out[b] = -s;
    }
}

extern "C" void kernel_launch(void* const* d_in, const int* in_sizes, int n_in,
                              void* d_out, int out_size, void* d_ws, size_t ws_size,
                              hipStream_t stream) {
    const float* x  = (const float*)d_in[0];
    const float* ap = (const float*)d_in[1];
    const float* aa = (const float*)d_in[2];
    float* out      = (float*)d_out;

    const int B = in_sizes[0] / (64 * 4);   // x is [B, 64, 4] floats

    simple_jastrow_wmma<<<B, 128, 0, stream>>>(x, ap, aa, out);
    (void)n_in; (void)out_size; (void)d_ws; (void)ws_size;
}